// MultiGCNInferenceNetwork_29643864277061
// MI455X (gfx1250) — compile-verified
//
#include <hip/hip_runtime.h>

#define NPG   1430   // NODES_PER_GRAPH (hard-coded in reference)
#define GENE0 1421   // first gene node index per graph
#define NHEADS 9

typedef __attribute__((ext_vector_type(2))) float v2f;
typedef __attribute__((ext_vector_type(8))) float v8f;

// ---------------------------------------------------------------- utilities
__global__ void k_zero(float* __restrict__ p, int n) {
  int i = blockIdx.x * blockDim.x + threadIdx.x;
  if (i < n) p[i] = 0.0f;
}

// deg[dst] += 1 per edge (hardware f32 atomic, L2-resident target)
__global__ void k_degree(const int* __restrict__ dst, float* __restrict__ deg, int E) {
  int i = blockIdx.x * blockDim.x + threadIdx.x;
  if (i < E) unsafeAtomicAdd(&deg[dst[i]], 1.0f);
}

// dinv = rsqrt(deg+1) (self-loop), fused with h1 = x @ W1  (W1 is 1x16)
__global__ void k_dinv_h1(const float* __restrict__ x, const float* __restrict__ W1,
                          float* __restrict__ degdinv, float* __restrict__ h1, int N) {
  int t = blockIdx.x * blockDim.x + threadIdx.x;
  if (t >= N * 16) return;
  int i = t >> 4, f = t & 15;
  h1[t] = x[i] * W1[f];
  if (f == 0) degdinv[i] = rsqrtf(degdinv[i] + 1.0f);
}

// agg[dst,f] += dinv[src]*dinv[dst] * h[src,f]; 16 lanes per edge so the
// h[src] gather and the atomic target are 64B-contiguous per edge.
__global__ void k_scatter(const int* __restrict__ src, const int* __restrict__ dst,
                          const float* __restrict__ dinv, const float* __restrict__ h,
                          float* __restrict__ agg, int tot) {
  int t = blockIdx.x * blockDim.x + threadIdx.x;
  if (t >= tot) return;
  int e = t >> 4, f = t & 15;
  int s = src[e], d = dst[e];
  float nrm = dinv[s] * dinv[d];
  unsafeAtomicAdd(&agg[d * 16 + f], nrm * h[s * 16 + f]);
}

// Fused: A = relu(agg1 + dinv^2*h1 + b1)  (layer-1 epilogue), then
// D = A @ W2 via 4x V_WMMA_F32_16X16X4_F32; writes h2pre in place over h1.
// One wave per 16-node tile; branch is wave-uniform -> EXEC all-ones at WMMA.
__global__ void k_relu_gemm16(const float* __restrict__ agg1, float* __restrict__ hbuf,
                              const float* __restrict__ dinv, const float* __restrict__ b1,
                              const float* __restrict__ W2, int ntiles) {
  int wid  = (blockIdx.x * blockDim.x + threadIdx.x) >> 5;
  int lane = threadIdx.x & 31;
  if (wid >= ntiles) return;
  int base = wid << 4;
  int hi = lane >> 4;      // half-wave selector (K pairs {0,1} vs {2,3})
  int r  = lane & 15;      // A-row / B-col within tile
  int arow = base + r;
  float di = dinv[arow];
  float d2 = di * di;

  v2f a[4], b[4];
#pragma unroll
  for (int c = 0; c < 4; ++c) {
    int f0 = 4 * c + 2 * hi;                       // ISA 16x4 A layout: vgpr j, half hi -> K = 4c + 2*hi + j
    float g0 = agg1[arow * 16 + f0]     + d2 * hbuf[arow * 16 + f0]     + b1[f0];
    float g1 = agg1[arow * 16 + f0 + 1] + d2 * hbuf[arow * 16 + f0 + 1] + b1[f0 + 1];
    a[c][0] = fmaxf(g0, 0.0f);
    a[c][1] = fmaxf(g1, 0.0f);
    b[c][0] = W2[f0 * 16 + r];                     // B 4x16: row K striped across lanes
    b[c][1] = W2[(f0 + 1) * 16 + r];
  }

  v8f acc = {0.f, 0.f, 0.f, 0.f, 0.f, 0.f, 0.f, 0.f};
#pragma unroll
  for (int c = 0; c < 4; ++c)
    acc = __builtin_amdgcn_wmma_f32_16x16x4_f32(false, a[c], false, b[c],
                                                (short)0, acc, false, false);

  // C/D layout: VGPR v -> row v (lanes 0-15) / row v+8 (lanes 16-31), col = lane&15
#pragma unroll
  for (int v = 0; v < 8; ++v)
    hbuf[(base + v + 8 * hi) * 16 + r] = acc[v];   // coalesced 64B row stores
}

// Layer-2 epilogue evaluated ONLY at gene nodes, fused with both head MLPs.
__global__ void k_heads(const float* __restrict__ agg2, const float* __restrict__ h2pre,
                        const float* __restrict__ dinv, const float* __restrict__ b2,
                        const float* __restrict__ fw1, const float* __restrict__ fb1,
                        const float* __restrict__ fw2, const float* __restrict__ fb2,
                        float* __restrict__ out, int G) {
  int t = blockIdx.x * blockDim.x + threadIdx.x;
  if (t >= NHEADS * G) return;
  int g = t / G, k = t - g * G;
  int node = GENE0 + g + k * NPG;
  float di = dinv[node];
  float d2 = di * di;
  float feat[16];
#pragma unroll
  for (int f = 0; f < 16; ++f)
    feat[f] = fmaxf(agg2[node * 16 + f] + d2 * h2pre[node * 16 + f] + b2[f], 0.0f);
  float pred = fb2[g];
#pragma unroll
  for (int h = 0; h < 8; ++h) {
    float z = fb1[g * 8 + h];
#pragma unroll
    for (int f = 0; f < 16; ++f) z += feat[f] * fw1[(g * 16 + f) * 8 + h];
    pred += fmaxf(z, 0.0f) * fw2[g * 8 + h];
  }
  out[g * G + k] = pred;
}

// ------------------------------------------------------------------ launcher
extern "C" void kernel_launch(void* const* d_in, const int* in_sizes, int n_in,
                              void* d_out, int out_size, void* d_ws, size_t ws_size,
                              hipStream_t stream) {
  const float* x   = (const float*)d_in[0];
  const int*   ei  = (const int*)  d_in[1];
  const float* b1  = (const float*)d_in[4];
  const float* W1  = (const float*)d_in[3];
  const float* W2  = (const float*)d_in[5];
  const float* b2  = (const float*)d_in[6];
  const float* fw1 = (const float*)d_in[7];
  const float* fb1 = (const float*)d_in[8];
  const float* fw2 = (const float*)d_in[9];
  const float* fb2 = (const float*)d_in[10];
  float* out = (float*)d_out;

  int N = in_sizes[0];
  int E = in_sizes[1] / 2;
  int G = in_sizes[2] / NHEADS;
  const int* src = ei;
  const int* dst = ei + E;

  // workspace: dinv[N] | bufA[N*16] (h1 -> h2pre in place) | bufB[N*16] (agg1 -> agg2)
  float* degdinv = (float*)d_ws;
  float* bufA = degdinv + N;
  float* bufB = bufA + (size_t)N * 16;

  const int B = 256;
  int n16    = N * 16;
  int tot    = E * 16;
  int ntiles = (N + 15) / 16;

  k_zero<<<(N + B - 1) / B, B, 0, stream>>>(degdinv, N);
  k_zero<<<(n16 + B - 1) / B, B, 0, stream>>>(bufB, n16);
  k_degree<<<(E + B - 1) / B, B, 0, stream>>>(dst, degdinv, E);
  k_dinv_h1<<<(n16 + B - 1) / B, B, 0, stream>>>(x, W1, degdinv, bufA, N);
  k_scatter<<<(tot + B - 1) / B, B, 0, stream>>>(src, dst, degdinv, bufA, bufB, tot);
  k_relu_gemm16<<<(ntiles * 32 + B - 1) / B, B, 0, stream>>>(bufB, bufA, degdinv, b1, W2, ntiles);
  k_zero<<<(n16 + B - 1) / B, B, 0, stream>>>(bufB, n16);
  k_scatter<<<(tot + B - 1) / B, B, 0, stream>>>(src, dst, degdinv, bufA, bufB, tot);
  int nh = NHEADS * G;
  k_heads<<<(nh + B - 1) / B, B, 0, stream>>>(bufB, bufA, degdinv, b2, fw1, fb1, fw2, fb2, out, G);
}